// HyperGraphGNN_73959336837228
// MI455X (gfx1250) — compile-verified
//
#include <hip/hip_runtime.h>
#include <hip/hip_bf16.h>

#define DM 12
#define NH 4
#define HD 48
#define NGRAPH 128

typedef float v2f __attribute__((ext_vector_type(2)));
typedef float v8f __attribute__((ext_vector_type(8)));

// ---------------------------------------------------------------------------
// float atomic max via integer monotone-mapping trick (sign-safe, no NaNs here)
// ---------------------------------------------------------------------------
__device__ __forceinline__ void atomic_max_f32(float* addr, float val) {
  if (val >= 0.0f) atomicMax((int*)addr, __float_as_int(val));
  else             atomicMin((unsigned int*)addr, __float_as_uint(val));
}

// ---------------------------------------------------------------------------
// Dense GEMM with WMMA:  Dst[M,N] = X[M,K] @ W[K,N] (+ bias[N])
// Row-major. M % (16*RT) == 0 (host guarantees), K compile-time (mult of 4).
// One wave handles RT stacked 16-row tiles x one 16-col tile; each B K-chunk
// is loaded once and reused across the RT WMMAs (branch-free inner loop,
// EXEC all-ones throughout as WMMA requires).
//   A 16x4: lanes 0-15 -> rows, K=kb+0,kb+1 in v0,v1; lanes 16-31 -> K=kb+2,kb+3.
//   B 4x16: lanes 0-15 -> col=lane, K=kb+0,kb+1; lanes 16-31 -> K=kb+2,kb+3.
//   C/D: vgpr r -> row r (lanes 0-15) / row r+8 (lanes 16-31), col = lane%16.
// ---------------------------------------------------------------------------
template <int K, int RT>
__global__ void k_gemm_wmma(const float* __restrict__ X, const float* __restrict__ W,
                            const float* __restrict__ bias, float* __restrict__ Dst,
                            int N, int row_tiles, int col_tiles) {
  int wave = (int)((blockIdx.x * blockDim.x + threadIdx.x) >> 5);
  int lane = (int)(threadIdx.x & 31);
  int sr_count = row_tiles / RT;                 // exact (host guarantees)
  if (wave >= sr_count * col_tiles) return;      // wave-uniform: EXEC stays full
  int sr = wave / col_tiles;
  int ct = wave - sr * col_tiles;
  int colbase = ct << 4;
  int half = lane >> 4;
  int l    = lane & 15;
  int col  = colbase + l;
  float valid = (col < N) ? 1.0f : 0.0f;         // branch-free B masking
  int colc    = (col < N) ? col : (N - 1);       // clamp: loads always in-bounds

  int row0 = (sr * RT) << 4;

  v8f acc[RT];
#pragma unroll
  for (int t = 0; t < RT; ++t) acc[t] = (v8f){0.f,0.f,0.f,0.f,0.f,0.f,0.f,0.f};

#pragma unroll
  for (int kb = 0; kb < K; kb += 4) {
    int krow = kb + half * 2;
    v2f b;
    b.x = W[(size_t)krow * N + colc] * valid;
    b.y = W[(size_t)(krow + 1) * N + colc] * valid;
#pragma unroll
    for (int t = 0; t < RT; ++t) {
      const float* xp = X + (size_t)(row0 + t * 16 + l) * K + krow;
      v2f a; a.x = xp[0]; a.y = xp[1];
      acc[t] = __builtin_amdgcn_wmma_f32_16x16x4_f32(false, a, false, b, (short)0,
                                                     acc[t], false, false);
    }
  }

  if (col < N) {
    float bb = bias ? bias[col] : 0.f;
#pragma unroll
    for (int t = 0; t < RT; ++t) {
      int rbase = row0 + t * 16 + half * 8;
#pragma unroll
      for (int r = 0; r < 8; ++r)
        Dst[(size_t)(rbase + r) * N + col] = acc[t][r] + bb;
    }
  }
}

// ---------------------------------------------------------------------------
// wa[which][k][h] = sum_c w[k, h*12+c] * att[h, c]   (12x4 per table, 3 tables)
// ---------------------------------------------------------------------------
__global__ void k_make_wa(const float* w_src, const float* att_src,
                          const float* w_dst, const float* att_dst,
                          const float* w_edge, const float* att_edge,
                          float* __restrict__ wa) {
  int t = threadIdx.x;           // 0..143
  if (t >= 144) return;
  int which = t / 48;
  int k = (t % 48) >> 2;
  int h = t & 3;
  const float* w   = (which == 0) ? w_src  : (which == 1) ? w_dst  : w_edge;
  const float* att = (which == 0) ? att_src: (which == 1) ? att_dst: att_edge;
  float s = 0.f;
  if (w && att) {
#pragma unroll
    for (int c = 0; c < DM; ++c) s += w[k * HD + h * DM + c] * att[h * DM + c];
  }
  wa[t] = s;
}

// alpha[i,h] = x[i,:] @ wa[:,h]
__global__ void k_alpha(const float* __restrict__ x, const float* __restrict__ wa,
                        float* __restrict__ alpha, int n) {
  int i = blockIdx.x * blockDim.x + threadIdx.x;
  if (i >= n) return;
  const float4* xv4 = (const float4*)(x + (size_t)i * DM);
  float xv[DM];
  float4 q0 = xv4[0], q1 = xv4[1], q2 = xv4[2];
  xv[0]=q0.x; xv[1]=q0.y; xv[2]=q0.z; xv[3]=q0.w;
  xv[4]=q1.x; xv[5]=q1.y; xv[6]=q1.z; xv[7]=q1.w;
  xv[8]=q2.x; xv[9]=q2.y; xv[10]=q2.z; xv[11]=q2.w;
  float4 out;
  float s0=0.f, s1=0.f, s2=0.f, s3=0.f;
#pragma unroll
  for (int k = 0; k < DM; ++k) {
    float v = xv[k];
    s0 += v * wa[k * NH + 0];
    s1 += v * wa[k * NH + 1];
    s2 += v * wa[k * NH + 2];
    s3 += v * wa[k * NH + 3];
  }
  out.x = s0; out.y = s1; out.z = s2; out.w = s3;
  *(float4*)(alpha + (size_t)i * NH) = out;
}

// init amax=-1e30, den=0, out = bias (per dst node)
__global__ void k_gat_init(float* __restrict__ amax, float* __restrict__ den,
                           float* __restrict__ out, const float* __restrict__ bias,
                           int ndst) {
  int i = blockIdx.x * blockDim.x + threadIdx.x;
  if (i >= ndst) return;
  float4 m = {-1e30f, -1e30f, -1e30f, -1e30f};
  float4 z = {0.f, 0.f, 0.f, 0.f};
  *(float4*)(amax + (size_t)i * NH) = m;
  *(float4*)(den  + (size_t)i * NH) = z;
#pragma unroll
  for (int c = 0; c < DM; ++c) out[(size_t)i * DM + c] = bias[c];
}

// pass A: a = leaky_relu(alpha_src[src] + alpha_dst[dst] + de@wa_e); atomicMax amax[dst]
__global__ void k_edge_a(const int* __restrict__ src, const int* __restrict__ dst,
                         const float* __restrict__ alpha_s, const float* __restrict__ alpha_d,
                         const float* __restrict__ de, const float* __restrict__ wa_e,
                         float* __restrict__ abuf, float* __restrict__ amax, int E) {
  int e = blockIdx.x * blockDim.x + threadIdx.x;
  if (e >= E) return;
  int s = src[e], d = dst[e];
  float ae0 = 0.f, ae1 = 0.f, ae2 = 0.f, ae3 = 0.f;
  if (de) {
    const float4* dv4 = (const float4*)(de + (size_t)e * DM);
    float dv[DM];
    float4 q0 = dv4[0], q1 = dv4[1], q2 = dv4[2];
    dv[0]=q0.x; dv[1]=q0.y; dv[2]=q0.z; dv[3]=q0.w;
    dv[4]=q1.x; dv[5]=q1.y; dv[6]=q1.z; dv[7]=q1.w;
    dv[8]=q2.x; dv[9]=q2.y; dv[10]=q2.z; dv[11]=q2.w;
#pragma unroll
    for (int k = 0; k < DM; ++k) {
      float v = dv[k];
      ae0 += v * wa_e[k * NH + 0];
      ae1 += v * wa_e[k * NH + 1];
      ae2 += v * wa_e[k * NH + 2];
      ae3 += v * wa_e[k * NH + 3];
    }
  }
  float4 as = *(const float4*)(alpha_s + (size_t)s * NH);
  float4 ad = *(const float4*)(alpha_d + (size_t)d * NH);
  float a0 = as.x + ad.x + ae0;
  float a1 = as.y + ad.y + ae1;
  float a2 = as.z + ad.z + ae2;
  float a3 = as.w + ad.w + ae3;
  a0 = (a0 >= 0.f) ? a0 : 0.2f * a0;
  a1 = (a1 >= 0.f) ? a1 : 0.2f * a1;
  a2 = (a2 >= 0.f) ? a2 : 0.2f * a2;
  a3 = (a3 >= 0.f) ? a3 : 0.2f * a3;
  float4 av = {a0, a1, a2, a3};
  *(float4*)(abuf + (size_t)e * NH) = av;
  float* am = amax + (size_t)d * NH;
  atomic_max_f32(am + 0, a0);
  atomic_max_f32(am + 1, a1);
  atomic_max_f32(am + 2, a2);
  atomic_max_f32(am + 3, a3);
}

// pass B: e = exp(a - amax[dst]); den[dst] += e
__global__ void k_edge_exp(const int* __restrict__ dst, float* __restrict__ abuf,
                           const float* __restrict__ amax, float* __restrict__ den, int E) {
  int e = blockIdx.x * blockDim.x + threadIdx.x;
  if (e >= E) return;
  int d = dst[e];
  float4 av = *(const float4*)(abuf + (size_t)e * NH);
  float4 mx = *(const float4*)(amax + (size_t)d * NH);
  float4 ev;
  ev.x = expf(av.x - mx.x);
  ev.y = expf(av.y - mx.y);
  ev.z = expf(av.z - mx.z);
  ev.w = expf(av.w - mx.w);
  *(float4*)(abuf + (size_t)e * NH) = ev;
  float* dn = den + (size_t)d * NH;
  atomicAdd(dn + 0, ev.x);
  atomicAdd(dn + 1, ev.y);
  atomicAdd(dn + 2, ev.z);
  atomicAdd(dn + 3, ev.w);
}

// pass C: w = e/(den+1e-16); out[dst,c] += 0.25 * sum_h w_h * hs[src, h*12+c]
__global__ void k_edge_scatter(const int* __restrict__ src, const int* __restrict__ dst,
                               const float* __restrict__ abuf, const float* __restrict__ den,
                               const float* __restrict__ hs, float* __restrict__ out, int E) {
  int e = blockIdx.x * blockDim.x + threadIdx.x;
  if (e >= E) return;
  int s = src[e], d = dst[e];
  float4 av = *(const float4*)(abuf + (size_t)e * NH);
  float4 dn = *(const float4*)(den + (size_t)d * NH);
  float w0 = 0.25f * av.x / (dn.x + 1e-16f);
  float w1 = 0.25f * av.y / (dn.y + 1e-16f);
  float w2 = 0.25f * av.z / (dn.z + 1e-16f);
  float w3 = 0.25f * av.w / (dn.w + 1e-16f);
  const float4* hr4 = (const float4*)(hs + (size_t)s * HD);
  float hr[HD];
#pragma unroll
  for (int q = 0; q < HD / 4; ++q) {
    float4 v = hr4[q];
    hr[4 * q + 0] = v.x; hr[4 * q + 1] = v.y; hr[4 * q + 2] = v.z; hr[4 * q + 3] = v.w;
  }
  float* op = out + (size_t)d * DM;
#pragma unroll
  for (int c = 0; c < DM; ++c) {
    float m = w0 * hr[c] + w1 * hr[DM + c] + w2 * hr[2 * DM + c] + w3 * hr[3 * DM + c];
    atomicAdd(op + c, m);
  }
}

// agg: out = LN(relu(concat(in0..in{nin-1}) @ W + b)) * g + beta
__global__ void k_agg(const float* __restrict__ in0, const float* __restrict__ in1,
                      const float* __restrict__ in2, const float* __restrict__ in3,
                      int nin, const float* __restrict__ Wm, const float* __restrict__ bvec,
                      const float* __restrict__ ln_g, const float* __restrict__ ln_b,
                      float* __restrict__ out, int n) {
  __shared__ float sW[HD * DM];
  __shared__ float sb[DM], sg[DM], sbe[DM];
  int K = nin * DM;
  for (int t = threadIdx.x; t < K * DM; t += blockDim.x) sW[t] = Wm[t];
  if (threadIdx.x < DM) {
    sb[threadIdx.x]  = bvec[threadIdx.x];
    sg[threadIdx.x]  = ln_g[threadIdx.x];
    sbe[threadIdx.x] = ln_b[threadIdx.x];
  }
  __syncthreads();
  int i = blockIdx.x * blockDim.x + threadIdx.x;
  if (i >= n) return;
  float xin[HD];
#pragma unroll
  for (int c = 0; c < DM; ++c) xin[c] = in0[(size_t)i * DM + c];
#pragma unroll
  for (int c = 0; c < DM; ++c) xin[DM + c] = in1[(size_t)i * DM + c];
  if (nin == 4) {
#pragma unroll
    for (int c = 0; c < DM; ++c) xin[2 * DM + c] = in2[(size_t)i * DM + c];
#pragma unroll
    for (int c = 0; c < DM; ++c) xin[3 * DM + c] = in3[(size_t)i * DM + c];
  }
  float h[DM];
  float mean = 0.f;
#pragma unroll
  for (int j = 0; j < DM; ++j) {
    float s = sb[j];
    for (int k = 0; k < K; ++k) s += xin[k] * sW[k * DM + j];
    s = (s > 0.f) ? s : 0.f;
    h[j] = s;
    mean += s;
  }
  mean *= (1.0f / DM);
  float var = 0.f;
#pragma unroll
  for (int j = 0; j < DM; ++j) { float dv = h[j] - mean; var += dv * dv; }
  var *= (1.0f / DM);
  float inv = rsqrtf(var + 1e-5f);
#pragma unroll
  for (int j = 0; j < DM; ++j)
    out[(size_t)i * DM + j] = (h[j] - mean) * inv * sg[j] + sbe[j];
}

__global__ void k_pool_init(float* __restrict__ gmax, float* __restrict__ gden,
                            float* __restrict__ pool) {
  int i = threadIdx.x;
  if (i >= NGRAPH) return;
  gmax[i] = -1e30f;
  gden[i] = 0.f;
#pragma unroll
  for (int c = 0; c < DM; ++c) pool[i * DM + c] = 0.f;
}

__global__ void k_gate_max(const float* __restrict__ x, const float* __restrict__ w,
                           const float* __restrict__ b, const int* __restrict__ batch,
                           float* __restrict__ gate, float* __restrict__ gmax, int n) {
  int i = blockIdx.x * blockDim.x + threadIdx.x;
  if (i >= n) return;
  float s = b[0];
#pragma unroll
  for (int k = 0; k < DM; ++k) s += x[(size_t)i * DM + k] * w[k];
  gate[i] = s;
  atomic_max_f32(&gmax[batch[i]], s);
}

__global__ void k_gate_exp(float* __restrict__ gate, const float* __restrict__ gmax,
                           float* __restrict__ gden, const int* __restrict__ batch, int n) {
  int i = blockIdx.x * blockDim.x + threadIdx.x;
  if (i >= n) return;
  int g = batch[i];
  float ev = expf(gate[i] - gmax[g]);
  gate[i] = ev;
  atomicAdd(&gden[g], ev);
}

__global__ void k_pool_acc(const float* __restrict__ x, const float* __restrict__ gate,
                           const float* __restrict__ gden, const int* __restrict__ batch,
                           float* __restrict__ pool, int n) {
  int i = blockIdx.x * blockDim.x + threadIdx.x;
  if (i >= n) return;
  int g = batch[i];
  float wv = gate[i] / (gden[g] + 1e-16f);
#pragma unroll
  for (int c = 0; c < DM; ++c)
    atomicAdd(&pool[g * DM + c], wv * x[(size_t)i * DM + c]);
}

__global__ void k_qhead(const float* __restrict__ xt, const int* __restrict__ batch_t,
                        const float* __restrict__ pt, const float* __restrict__ pr,
                        const float* __restrict__ w1, const float* __restrict__ b1,
                        const float* __restrict__ w2, const float* __restrict__ b2,
                        const float* __restrict__ w3, const float* __restrict__ b3,
                        float* __restrict__ out, int n) {
  __shared__ float sw1[36 * DM];
  __shared__ float sw2[DM * 6];
  __shared__ float sw3[6], sb1[DM], sb2[6], sb3v[1];
  for (int t = threadIdx.x; t < 36 * DM; t += blockDim.x) sw1[t] = w1[t];
  for (int t = threadIdx.x; t < DM * 6; t += blockDim.x) sw2[t] = w2[t];
  if (threadIdx.x < 6)  sw3[threadIdx.x] = w3[threadIdx.x];
  if (threadIdx.x < DM) sb1[threadIdx.x] = b1[threadIdx.x];
  if (threadIdx.x < 6)  sb2[threadIdx.x] = b2[threadIdx.x];
  if (threadIdx.x == 0) sb3v[0] = b3[0];
  __syncthreads();
  int i = blockIdx.x * blockDim.x + threadIdx.x;
  if (i >= n) return;
  int g = batch_t[i];
  float inp[36];
#pragma unroll
  for (int c = 0; c < DM; ++c) inp[c] = xt[(size_t)i * DM + c];
#pragma unroll
  for (int c = 0; c < DM; ++c) inp[DM + c] = pt[g * DM + c];
#pragma unroll
  for (int c = 0; c < DM; ++c) inp[2 * DM + c] = pr[g * DM + c];
  float h1[DM];
#pragma unroll
  for (int j = 0; j < DM; ++j) {
    float s = sb1[j];
    for (int k = 0; k < 36; ++k) s += inp[k] * sw1[k * DM + j];
    h1[j] = (s > 0.f) ? s : 0.f;
  }
  float h2[6];
#pragma unroll
  for (int j = 0; j < 6; ++j) {
    float s = sb2[j];
    for (int k = 0; k < DM; ++k) s += h1[k] * sw2[k * 6 + j];
    h2[j] = (s > 0.f) ? s : 0.f;
  }
  float o = sb3v[0];
#pragma unroll
  for (int k = 0; k < 6; ++k) o += h2[k] * sw3[k];
  out[i] = o;
}

// ---------------------------------------------------------------------------
// Host side
// ---------------------------------------------------------------------------
static inline int cdiv(long long a, long long b) { return (int)((a + b - 1) / b); }

// jax pytree leaf order (dicts sorted by key, lists in order): 76 leaves.
static const int g_leaf_sizes[76] = {
  /* dem_exp.b,w */ 12, 48,
  /* layer 0 */ 12,12,12,288, 12,12,12,576,
                48,48,12,576,576,  48,48,12,576,576,  48,48,12,576,576,
                48,48,48,12,576,576,576,
  /* layer 1 */ 12,12,12,288, 12,12,12,576,
                48,48,12,576,576,  48,48,12,576,576,  48,48,12,576,576,
                48,48,48,12,576,576,576,
  /* q: b1,b2,b3,w1,w2,w3 */ 12, 6, 1, 432, 72, 6,
  /* res_exp.b,w */ 12, 96,
  /* res_pool.b,w */ 1, 12,
  /* task_exp.b,w */ 12, 192,
  /* task_pool.b,w */ 1, 12
};

extern "C" void kernel_launch(void* const* d_in, const int* in_sizes, int n_in,
                              void* d_out, int out_size, void* d_ws, size_t ws_size,
                              hipStream_t stream) {
  const float* x_tasks = (const float*)d_in[0];
  const float* x_res   = (const float*)d_in[1];
  const float* dem     = (const float*)d_in[2];
  const int* dsrc = (const int*)d_in[3];
  const int* ddst = (const int*)d_in[4];
  const int* psrc = (const int*)d_in[5];
  const int* pdst = (const int*)d_in[6];
  const int* ssrc = (const int*)d_in[7];
  const int* sdst = (const int*)d_in[8];
  const int* rsrc = (const int*)d_in[9];
  const int* rdst = (const int*)d_in[10];
  const int* batch_t = (const int*)d_in[11];
  const int* batch_r = (const int*)d_in[12];

  const int NT  = in_sizes[0] / 16;
  const int NR  = in_sizes[1] / 8;
  const int ED  = in_sizes[2] / 4;
  const int EP  = in_sizes[5];
  const int ES  = in_sizes[7];
  const int ERD = in_sizes[9];
  int Emax = EP; if (ES > Emax) Emax = ES; if (ED > Emax) Emax = ED; if (ERD > Emax) Emax = ERD;

  // Parameter leaf pointers (separate leaves or one concatenated buffer).
  const float* P[76];
  if (n_in >= 13 + 76) {
    for (int i = 0; i < 76; ++i) P[i] = (const float*)d_in[13 + i];
  } else {
    const float* base = (const float*)d_in[13];
    size_t off = 0;
    for (int i = 0; i < 76; ++i) { P[i] = base + off; off += (size_t)g_leaf_sizes[i]; }
  }

  // Workspace carve-up (floats; every block a multiple of 4 floats -> 16B aligned).
  float* W = (float*)d_ws;
  size_t off = 0;
  auto alloc = [&](size_t nf) { float* p = W + off; off += nf; return p; };
  float* xt[2]   = { alloc((size_t)NT * DM), alloc((size_t)NT * DM) };
  float* xr[2]   = { alloc((size_t)NR * DM), alloc((size_t)NR * DM) };
  float* deb     = alloc((size_t)ED * DM);
  float* hs      = alloc((size_t)NT * HD);
  float* alpha_s = alloc((size_t)NT * NH);
  float* alpha_d = alloc((size_t)NT * NH);
  float* abuf    = alloc((size_t)Emax * NH);
  float* amax    = alloc((size_t)NT * NH);
  float* den     = alloc((size_t)NT * NH);
  float* xrt     = alloc((size_t)NR * DM);
  float* xp      = alloc((size_t)NT * DM);
  float* xs      = alloc((size_t)NT * DM);
  float* xrr     = alloc((size_t)NT * DM);
  float* gate    = alloc((size_t)NT);
  float* gmax    = alloc(NGRAPH);
  float* gden    = alloc(NGRAPH);
  float* ptb     = alloc(NGRAPH * DM);
  float* prb     = alloc(NGRAPH * DM);
  float* wa      = alloc(144);
  (void)ws_size;

  const int TB = 256;
  auto gemm = [&](const float* X, const float* Wm, const float* bias, float* Dst,
                  int M, int K, int N) {
    int rt = M / 16;
    int ct = (N + 15) / 16;
    bool r4 = (rt % 4) == 0;
    int waves = (r4 ? rt / 4 : rt) * ct;
    int blocks = cdiv(waves, TB / 32);
    switch (K) {
      case 4:
        if (r4) k_gemm_wmma<4, 4><<<blocks, TB, 0, stream>>>(X, Wm, bias, Dst, N, rt, ct);
        else    k_gemm_wmma<4, 1><<<blocks, TB, 0, stream>>>(X, Wm, bias, Dst, N, rt, ct);
        break;
      case 8:
        if (r4) k_gemm_wmma<8, 4><<<blocks, TB, 0, stream>>>(X, Wm, bias, Dst, N, rt, ct);
        else    k_gemm_wmma<8, 1><<<blocks, TB, 0, stream>>>(X, Wm, bias, Dst, N, rt, ct);
        break;
      case 12:
        if (r4) k_gemm_wmma<12, 4><<<blocks, TB, 0, stream>>>(X, Wm, bias, Dst, N, rt, ct);
        else    k_gemm_wmma<12, 1><<<blocks, TB, 0, stream>>>(X, Wm, bias, Dst, N, rt, ct);
        break;
      default:
        if (r4) k_gemm_wmma<16, 4><<<blocks, TB, 0, stream>>>(X, Wm, bias, Dst, N, rt, ct);
        else    k_gemm_wmma<16, 1><<<blocks, TB, 0, stream>>>(X, Wm, bias, Dst, N, rt, ct);
        break;
    }
  };

  auto run_gat = [&](const float* xsrc, int nsrc, const float* xdst, int ndst,
                     const int* esrc, const int* edst, int E, const float* de_feat,
                     const float* w_src, const float* att_src,
                     const float* w_dst, const float* att_dst,
                     const float* w_edge, const float* att_edge,
                     const float* bias, float* outbuf) {
    gemm(xsrc, w_src, nullptr, hs, nsrc, DM, HD);
    k_make_wa<<<1, 144, 0, stream>>>(w_src, att_src, w_dst, att_dst, w_edge, att_edge, wa);
    k_alpha<<<cdiv(nsrc, TB), TB, 0, stream>>>(xsrc, wa, alpha_s, nsrc);
    k_alpha<<<cdiv(ndst, TB), TB, 0, stream>>>(xdst, wa + 48, alpha_d, ndst);
    k_gat_init<<<cdiv(ndst, TB), TB, 0, stream>>>(amax, den, outbuf, bias, ndst);
    k_edge_a<<<cdiv(E, TB), TB, 0, stream>>>(esrc, edst, alpha_s, alpha_d, de_feat,
                                             wa + 96, abuf, amax, E);
    k_edge_exp<<<cdiv(E, TB), TB, 0, stream>>>(edst, abuf, amax, den, E);
    k_edge_scatter<<<cdiv(E, TB), TB, 0, stream>>>(esrc, edst, abuf, den, hs, outbuf, E);
  };

  // Input expansions (WMMA GEMMs with fused bias).
  gemm(x_tasks, P[73], P[72], xt[0], NT, 16, DM);
  gemm(x_res,   P[69], P[68], xr[0], NR, 8,  DM);
  gemm(dem,     P[1],  P[0],  deb,   ED, 4,  DM);

  int cur = 0;
  for (int l = 0; l < 2; ++l) {
    const int LB = 2 + l * 30;
    // gat_tfr: tasks -> resources, with edge attrs
    run_gat(xt[cur], NT, xr[cur], NR, dsrc, ddst, ED, deb,
            P[LB + 29], P[LB + 25], P[LB + 27], P[LB + 23],
            P[LB + 28], P[LB + 24], P[LB + 26], xrt);
    // agg_res: xr_new = LN(relu([xr, xrt] @ W + b))
    k_agg<<<cdiv(NR, TB), TB, 0, stream>>>(xr[cur], xrt, nullptr, nullptr, 2,
                                           P[LB + 3], P[LB + 0], P[LB + 2], P[LB + 1],
                                           xr[1 - cur], NR);
    // gat_pred: tasks -> tasks
    run_gat(xt[cur], NT, xt[cur], NT, psrc, pdst, EP, nullptr,
            P[LB + 12], P[LB + 9], P[LB + 11], P[LB + 8],
            nullptr, nullptr, P[LB + 10], xp);
    // gat_succ: tasks -> tasks
    run_gat(xt[cur], NT, xt[cur], NT, ssrc, sdst, ES, nullptr,
            P[LB + 22], P[LB + 19], P[LB + 21], P[LB + 18],
            nullptr, nullptr, P[LB + 20], xs);
    // gat_rft: (updated) resources -> tasks
    run_gat(xr[1 - cur], NR, xt[cur], NT, rsrc, rdst, ERD, nullptr,
            P[LB + 17], P[LB + 14], P[LB + 16], P[LB + 13],
            nullptr, nullptr, P[LB + 15], xrr);
    // agg_tasks: xt_new = LN(relu([xt, xp, xs, xrr] @ W + b))
    k_agg<<<cdiv(NT, TB), TB, 0, stream>>>(xt[cur], xp, xs, xrr, 4,
                                           P[LB + 7], P[LB + 4], P[LB + 6], P[LB + 5],
                                           xt[1 - cur], NT);
    cur = 1 - cur;
  }

  // Attentional pooling (tasks).
  k_pool_init<<<1, NGRAPH, 0, stream>>>(gmax, gden, ptb);
  k_gate_max<<<cdiv(NT, TB), TB, 0, stream>>>(xt[cur], P[75], P[74], batch_t, gate, gmax, NT);
  k_gate_exp<<<cdiv(NT, TB), TB, 0, stream>>>(gate, gmax, gden, batch_t, NT);
  k_pool_acc<<<cdiv(NT, TB), TB, 0, stream>>>(xt[cur], gate, gden, batch_t, ptb, NT);
  // Attentional pooling (resources).
  k_pool_init<<<1, NGRAPH, 0, stream>>>(gmax, gden, prb);
  k_gate_max<<<cdiv(NR, TB), TB, 0, stream>>>(xr[cur], P[71], P[70], batch_r, gate, gmax, NR);
  k_gate_exp<<<cdiv(NR, TB), TB, 0, stream>>>(gate, gmax, gden, batch_r, NR);
  k_pool_acc<<<cdiv(NR, TB), TB, 0, stream>>>(xr[cur], gate, gden, batch_r, prb, NR);

  // Q head.
  k_qhead<<<cdiv(NT, TB), TB, 0, stream>>>(xt[cur], batch_t, ptb, prb,
                                           P[65], P[62], P[66], P[63], P[67], P[64],
                                           (float*)d_out, NT);
  (void)out_size;
}